// HawkesPointProcess_23063974380162
// MI455X (gfx1250) — compile-verified
//
#include <hip/hip_runtime.h>
#include <hip/hip_bf16.h>

typedef __attribute__((ext_vector_type(2))) float v2f;
typedef __attribute__((ext_vector_type(8))) float v8f;

static constexpr int kT      = 2048;
static constexpr int kTile   = 16;
static constexpr int kNTiles = kT / kTile;   // 128
static constexpr int kBlock  = 256;          // 8 waves (wave32)
static constexpr int kNWaves = kBlock / 32;

__device__ __forceinline__ float softplusf(float x) {
    return (x > 20.f) ? x : log1pf(expf(x));
}

__global__ __launch_bounds__(kBlock)
void HawkesPointProcess_23063974380162_kernel(const float* __restrict__ tin,
                                              const float* __restrict__ mu_p,
                                              const float* __restrict__ alpha_p,
                                              const float* __restrict__ beta_p,
                                              float* __restrict__ out, int N)
{
    __shared__ float ts[kT];        // event times for this sequence
    __shared__ float sdiag[kT];     // diagonal-tile causal sums; reused as lamb
    __shared__ float Vs[kNTiles];   // per-tile sum of exp(beta*(t_j - e_I))
    __shared__ float es[kNTiles];   // per-tile end time e_I
    __shared__ float Ps[kNTiles];   // prefix through tile I, relative to e_I
    __shared__ float red[kNWaves];

    const int n    = blockIdx.x;
    const int tid  = threadIdx.x;
    const int lane = tid & 31;
    const int wave = tid >> 5;

    const float mu_s = softplusf(mu_p[0]);
    const float al_s = softplusf(alpha_p[0]);
    const float bs   = softplusf(beta_p[0]);

    // ---- Phase 0: stage t into LDS (b128 loads) ----
    const float4* tg  = reinterpret_cast<const float4*>(tin + (size_t)n * kT);
    float4*       ts4 = reinterpret_cast<float4*>(ts);
    for (int i = tid; i < kT / 4; i += kBlock) ts4[i] = tg[i];
    __syncthreads();

    // ---- Phase 1: WMMA row-sums of masked exp matrix on each 16x16 diagonal tile ----
    // A layout (32-bit A 16x4): lanes 0-15 hold K=0 (VGPR0) / K=1 (VGPR1) for row M=lane;
    // lanes 16-31 hold K=2/K=3 for row M=lane-16. B = all-ones is layout-invariant.
    const v2f ones = {1.0f, 1.0f};
    const int ml = lane & 15;
    const int cb = (lane >> 4) << 1;   // 0 for low half-wave, 2 for high half-wave

    for (int I = wave; I < kNTiles; I += kNWaves) {
        const int base = I * kTile;
        const float tm = ts[base + ml];
        v8f c = {0.f, 0.f, 0.f, 0.f, 0.f, 0.f, 0.f, 0.f};
#pragma unroll
        for (int k0 = 0; k0 < kTile; k0 += 4) {
            const int c0 = k0 + cb;                       // even -> 8B aligned
            const float2 tc = *reinterpret_cast<const float2*>(&ts[base + c0]);
            // Always-live exp chain; causal mask applied AFTER via single cndmask.
            const float ex = __expf((tc.x - tm) * bs);
            const float ey = __expf((tc.y - tm) * bs);
            v2f a;
            a.x = (c0     < ml) ? ex : 0.f;
            a.y = (c0 + 1 < ml) ? ey : 0.f;
            c = __builtin_amdgcn_wmma_f32_16x16x4_f32(false, a, false, ones,
                                                      (short)0, c, false, false);
        }
        // D layout: lanes 0-15 hold rows 0-7 in c[0..7]; lanes 16-31 hold rows 8-15.
        // Tile aggregate for the prefix scan comes FREE from the WMMA result:
        //   V_I = sum_j exp(bs*(t_j - t15)) = 1 + S_diag[row 15] = 1 + c[7] (high half).
        if (ml == 0) {
            const int half = lane >> 4;
            const int o = base + half * 8;
#pragma unroll
            for (int v = 0; v < 8; ++v) sdiag[o + v] = c[v];
            if (half) {
                Vs[I] = c[7] + 1.0f;
                es[I] = ts[base + 15];
            }
        }
    }
    __syncthreads();

    // ---- Phase 2: sequential prefix over 128 tile aggregates (cheap) ----
    if (tid == 0) {
        float P = 0.f, eprev = es[0];
        for (int I = 0; I < kNTiles; ++I) {
            P = P * __expf(bs * (eprev - es[I])) + Vs[I];
            Ps[I] = P;
            eprev = es[I];
        }
    }
    __syncthreads();

    // ---- Phase 3: lamb, loglik, compensator, dist ----
    // e_{127} == t_last, so Ps[127] == sum_j exp(-beta*(t_last - t_j)) exactly.
    const float tlast = ts[kT - 1];
    const float comp  = (tlast - ts[0]) * mu_s - (al_s / bs) * Ps[kNTiles - 1];

    float* out_ll   = out;                                      // (N,)
    float* out_dist = out + N + (size_t)n * kT;                 // (N,T)
    float* out_lamb = out + N + (size_t)N * kT + (size_t)n * kT;// (N,T)

    float part = 0.f;
    for (int i = tid; i < kT; i += kBlock) {
        const int I  = i >> 4;
        const int Ip = (I > 0) ? I - 1 : 0;          // clamped; result masked below
        const float sp = Ps[Ip] * __expf(-bs * (ts[i] - es[Ip]));
        const float S  = sdiag[i] + ((I > 0) ? sp : 0.f);
        const float lam = mu_s + al_s * S;
        out_lamb[i] = lam;
        sdiag[i] = lam;                 // exclusive slot per thread; reuse as lamb
        part += logf(lam + 1e-8f);
    }
#pragma unroll
    for (int off = 16; off >= 1; off >>= 1) part += __shfl_xor(part, off, 32);
    if (lane == 0) red[wave] = part;
    __syncthreads();
    if (tid == 0) {
        float s = 0.f;
        for (int w = 0; w < kNWaves; ++w) s += red[w];
        red[0] = s;
    }
    __syncthreads();

    if (tid == 0) out_ll[n] = red[0] - comp;

    const float ec = __expf(-comp);
    for (int i = tid; i < kT; i += kBlock) out_dist[i] = sdiag[i] * ec;
}

extern "C" void kernel_launch(void* const* d_in, const int* in_sizes, int n_in,
                              void* d_out, int out_size, void* d_ws, size_t ws_size,
                              hipStream_t stream) {
    const float* tin   = (const float*)d_in[0];  // (N, T, 1) float32
    const float* mu    = (const float*)d_in[1];
    const float* alpha = (const float*)d_in[2];
    const float* beta  = (const float*)d_in[3];
    float* out = (float*)d_out;
    const int N = in_sizes[0] / kT;
    hipLaunchKernelGGL(HawkesPointProcess_23063974380162_kernel,
                       dim3(N), dim3(kBlock), 0, stream,
                       tin, mu, alpha, beta, out, N);
}